// NNSystem_mech_22187801051469
// MI455X (gfx1250) — compile-verified
//
#include <hip/hip_runtime.h>
#include <math.h>

// ---------------------------------------------------------------------------
// HPA-axis neural-ODE RHS on MI455X (gfx1250), single wave32.
//
// The four ANN(1,10,1) output dot-products are computed as one 4x10 @ 10x4
// matmul via three chained V_WMMA_F32_16X16X4_F32 (K=4 each, 12 >= 10),
// taking the diagonal of D. A/B operands are built directly in the ISA's
// 32-bit 16x4 / 4x16 VGPR layouts so no pre-WMMA shuffles are needed.
// ---------------------------------------------------------------------------

typedef __attribute__((ext_vector_type(2))) float v2f;
typedef __attribute__((ext_vector_type(8))) float v8f;

struct HpaParams {
  const float* y;
  const float* aiw; const float* aib;
  const float* apw1; const float* apb1; const float* apw2; const float* apb2;
  const float* anw1; const float* anb1; const float* anw2; const float* anb2;
  const float* arw;  const float* arb;
  const float* ciw; const float* cib;
  const float* cpw1; const float* cpb1; const float* cpw2; const float* cpb2;
  const float* cnw1; const float* cnb1; const float* cnw2; const float* cnb2;
  const float* crw;  const float* crb;
  const float* K_i;  const float* n_hill;
  float* out;
};

__device__ __forceinline__ float sp(float x) {
  // overflow-safe softplus: max(x,0) + log1p(exp(-|x|))
  return fmaxf(x, 0.0f) + log1pf(expf(-fabsf(x)));
}

__global__ __launch_bounds__(32) void hpa_ode_rhs_kernel(HpaParams p) {
  const int lane  = (int)threadIdx.x;   // 0..31, one full wave32
  const int n     = lane & 15;          // matrix row/col: MLP index when < 4
  const int khalf = lane >> 4;          // 0: K={0,1}, 1: K={2,3} per WMMA

  const float y0 = p.y[0];
  const float y1 = p.y[1];

  // initial 1->2 linears + softplus (redundant on all lanes; trivial cost)
  const float a0 = sp(fmaf(p.aiw[0], y0, p.aib[0]));
  const float a1 = sp(fmaf(p.aiw[1], y0, p.aib[1]));
  const float c0 = sp(fmaf(p.ciw[0], y1, p.cib[0]));
  const float c1 = sp(fmaf(p.ciw[1], y1, p.cib[1]));

  // Per-lane MLP selection (branch-free selects; EXEC must stay all-1s).
  // MLP 0: acth_pos(a0)  1: acth_neg(a1)  2: cort_pos(c0)  3: cort_neg(c1)
  const bool  valid = (n < 4);
  const float* w1 = (n == 1) ? p.anw1 : (n == 2) ? p.cpw1 : (n == 3) ? p.cnw1 : p.apw1;
  const float* b1 = (n == 1) ? p.anb1 : (n == 2) ? p.cpb1 : (n == 3) ? p.cnb1 : p.apb1;
  const float* w2 = (n == 1) ? p.anw2 : (n == 2) ? p.cpw2 : (n == 3) ? p.cnw2 : p.apw2;
  const float  x  = (n == 1) ? a1     : (n == 2) ? c0     : (n == 3) ? c1     : a0;

  // D[m][n] = sum_j w2_m[j] * softplus(w1_n[j]*x_n + b1_n[j]); want diag m==n.
  v8f acc = {};
#pragma unroll
  for (int g = 0; g < 3; ++g) {               // K chunks: j = 4g .. 4g+3
    const int  j0   = 4 * g + 2 * khalf;
    const int  j1   = j0 + 1;
    const bool ok0  = valid && (j0 < 10);
    const bool ok1  = valid && (j1 < 10);
    const int  j0c  = (j0 < 10) ? j0 : 9;     // clamp so every lane loads
    const int  j1c  = (j1 < 10) ? j1 : 9;     // valid addresses (masked below)

    v2f aV, bV;                                // A: 16x4 f32, B: 4x16 f32
    aV.x = ok0 ? w2[j0c] : 0.0f;
    aV.y = ok1 ? w2[j1c] : 0.0f;
    bV.x = ok0 ? sp(fmaf(w1[j0c], x, b1[j0c])) : 0.0f;
    bV.y = ok1 ? sp(fmaf(w1[j1c], x, b1[j1c])) : 0.0f;

    // D = A x B + C  (8 args: neg_a, A, neg_b, B, c_mod, C, reuse_a, reuse_b)
    acc = __builtin_amdgcn_wmma_f32_16x16x4_f32(
        false, aV, false, bV, (short)0, acc, false, false);
  }

  // Diagonal: D[m][m] lives in accumulator VGPR m, lane m (m < 8, lane < 16).
  const float d0 = __shfl(acc[0], 0, 32);
  const float d1 = __shfl(acc[1], 1, 32);
  const float d2 = __shfl(acc[2], 2, 32);
  const float d3 = __shfl(acc[3], 3, 32);

  const float a_pos = d0 + p.apb2[0];
  const float a_neg = d1 + p.anb2[0];
  const float c_pos = d2 + p.cpb2[0];
  const float c_neg = d3 + p.cnb2[0];

  // Hill inhibition of ACTH positive term by cortisol
  const float nh   = p.n_hill[0];
  const float Kn   = powf(p.K_i[0], nh);
  const float hill = Kn / (Kn + powf(y1, nh));

  const float dy_acth =
      fmaf(p.arw[0], sp(hill * a_pos), fmaf(p.arw[1], sp(a_neg), p.arb[0]));
  const float dy_cort =
      fmaf(p.crw[0], sp(y0 * c_pos), fmaf(p.crw[1], sp(c_neg), p.crb[0]));

  if (lane == 0) {
    p.out[0] = dy_acth;
    p.out[1] = dy_cort;
  }
}

extern "C" void kernel_launch(void* const* d_in, const int* in_sizes, int n_in,
                              void* d_out, int out_size, void* d_ws, size_t ws_size,
                              hipStream_t stream) {
  (void)in_sizes; (void)n_in; (void)out_size; (void)d_ws; (void)ws_size;

  HpaParams p;
  // d_in[0] is t (unused by the RHS).
  p.y    = (const float*)d_in[1];
  p.aiw  = (const float*)d_in[2];
  p.aib  = (const float*)d_in[3];
  p.apw1 = (const float*)d_in[4];
  p.apb1 = (const float*)d_in[5];
  p.apw2 = (const float*)d_in[6];
  p.apb2 = (const float*)d_in[7];
  p.anw1 = (const float*)d_in[8];
  p.anb1 = (const float*)d_in[9];
  p.anw2 = (const float*)d_in[10];
  p.anb2 = (const float*)d_in[11];
  p.arw  = (const float*)d_in[12];
  p.arb  = (const float*)d_in[13];
  p.ciw  = (const float*)d_in[14];
  p.cib  = (const float*)d_in[15];
  p.cpw1 = (const float*)d_in[16];
  p.cpb1 = (const float*)d_in[17];
  p.cpw2 = (const float*)d_in[18];
  p.cpb2 = (const float*)d_in[19];
  p.cnw1 = (const float*)d_in[20];
  p.cnb1 = (const float*)d_in[21];
  p.cnw2 = (const float*)d_in[22];
  p.cnb2 = (const float*)d_in[23];
  p.crw  = (const float*)d_in[24];
  p.crb  = (const float*)d_in[25];
  p.K_i  = (const float*)d_in[26];
  p.n_hill = (const float*)d_in[27];
  p.out  = (float*)d_out;

  hpa_ode_rhs_kernel<<<dim3(1), dim3(32), 0, stream>>>(p);
}